// expSumLayerWindow_59906203845038
// MI455X (gfx1250) — compile-verified
//
#include <hip/hip_runtime.h>
#include <hip/hip_bf16.h>

// CDNA5 (gfx1250) wave32 WMMA types
typedef __attribute__((ext_vector_type(2))) float v2f;   // A/B for V_WMMA_F32_16X16X4_F32 (16x4 f32 = 2 VGPRs/lane)
typedef __attribute__((ext_vector_type(8))) float v8f;   // C/D 16x16 f32 accumulator

#define KN 2048          // N (points per batch row)
#define MU 2.0f
#define INV_WTRANS 2.0f  // 1 / 0.5
#define WWIDTH 3.0f

// erfc(t) for t >= 0 via Abramowitz-Stegun 7.1.26: poly(1/(1+p t)) * exp(-t^2)
__device__ __forceinline__ float erfc_pos(float t) {
    const float p = 0.3275911f;
    float s = __builtin_amdgcn_rcpf(__builtin_fmaf(p, t, 1.0f));
    float poly = s * (0.254829592f +
                 s * (-0.284496736f +
                 s * (1.421413741f +
                 s * (-1.453152027f +
                 s * 1.061405429f))));
    return poly * __expf(-t * t);
}

__device__ __forceinline__ float erfc_full(float t) {
    float a = fabsf(t);
    float e = erfc_pos(a);
    return (t >= 0.0f) ? e : (2.0f - e);
}

// kernel(x) = |std| * ( (1 - 0.5*erfc(x/winTrans - winWidth)) * exp(-mu*x) - bias )
__device__ __forceinline__ float pair_kernel(float x, float astd, float bias) {
    float t = __builtin_fmaf(INV_WTRANS, x, -WWIDTH);
    float w = __builtin_fmaf(-0.5f, erfc_full(t), 1.0f);
    float e = __expf(-MU * x);
    return astd * (__builtin_fmaf(w, e, -bias));
}

__global__ void __launch_bounds__(256)
expSumLayerWindow_59906203845038_kernel(const float* __restrict__ Rin,
                                        const float* __restrict__ stdp,
                                        const float* __restrict__ biasp,
                                        float* __restrict__ out) {
    __shared__ float sR[KN];   // one 8KB batch row, shared by the whole workgroup

    const int batch = blockIdx.y;                 // 0..15
    const int tid   = threadIdx.x;                // 0..255 (8 wave32 waves)
    const int lane  = tid & 31;
    const int wave  = tid >> 5;

    // Cooperatively stage R[batch, :] into LDS (2048 floats / 256 threads = 8 each)
    const float* rowg = Rin + (size_t)batch * KN;
    #pragma unroll
    for (int k = tid; k < KN; k += 256) sR[k] = rowg[k];
    __syncthreads();

    const float astd = fabsf(stdp[0]);
    const float bias = biasp[0];

    // This wave owns i-rows [i0, i0+16)
    const int i0   = blockIdx.x * 128 + wave * 16;
    const int m    = lane & 15;      // matrix row this lane contributes (A-matrix M = lane % 16)
    const int half = lane >> 4;      // lanes 0-15 -> K = {0,1}; lanes 16-31 -> K = {2,3}
    const float ri = sR[i0 + m];

    v8f acc = {};                    // 16x16 f32 accumulator: every column = running row-sum
    const v2f ones = {1.0f, 1.0f};   // B matrix (4x16 of ones) -> WMMA performs the j-reduction

    // Sweep all j in steps of K=4; each lane evaluates 2 pair kernels per step.
    #pragma unroll 4
    for (int j0 = 0; j0 < KN; j0 += 4) {
        const int jb = j0 + (half << 1);
        v2f rj = *(const v2f*)(sR + jb);          // ds_load_b64, broadcast within half-wave

        v2f a;
        a.x = pair_kernel(fabsf(rj.x - ri), astd, bias);
        a.y = pair_kernel(fabsf(rj.y - ri), astd, bias);

        // acc[m,n] += sum_k A[m,k] * 1  -> row-sum accumulation on the matrix pipe
        acc = __builtin_amdgcn_wmma_f32_16x16x4_f32(
                  /*neg_a=*/false, a,
                  /*neg_b=*/false, ones,
                  /*c_mod=*/(short)0, acc,
                  /*reuse_a=*/false, /*reuse_b=*/false);
    }

    // Remove the i==j self-pair (x = 0) contribution, added once per row by the full sweep.
    const float diag = pair_kernel(0.0f, astd, bias);

    // C/D layout: VGPR r holds row m=r on lanes 0-15 and m=r+8 on lanes 16-31;
    // all 16 columns are identical, so lane 0 / lane 16 each carry 8 row-sums.
    float* orow = out + (size_t)batch * KN + i0;
    if (lane == 0) {
        #pragma unroll
        for (int r = 0; r < 8; ++r) orow[r] = acc[r] - diag;
    } else if (lane == 16) {
        #pragma unroll
        for (int r = 0; r < 8; ++r) orow[8 + r] = acc[r] - diag;
    }
}

extern "C" void kernel_launch(void* const* d_in, const int* in_sizes, int n_in,
                              void* d_out, int out_size, void* d_ws, size_t ws_size,
                              hipStream_t stream) {
    const float* Rin  = (const float*)d_in[0];
    const float* stdp = (const float*)d_in[1];
    const float* biasp= (const float*)d_in[2];
    float* out = (float*)d_out;

    const int B = in_sizes[0] / KN;              // 16
    dim3 grid(KN / 128, B);                      // x: i-chunks of 128 (8 waves x 16 rows), y: batch
    dim3 block(256);
    expSumLayerWindow_59906203845038_kernel<<<grid, block, 0, stream>>>(Rin, stdp, biasp, out);
}